// GATModel_18708877542133
// MI455X (gfx1250) — compile-verified
//
#include <hip/hip_runtime.h>
#include <math.h>

#define NN 100000
#define EE 1600000
#define GG 128
#define NEG_SLOPE 0.2f

typedef __attribute__((ext_vector_type(2))) float v2f;
typedef __attribute__((ext_vector_type(8))) float v8f;

// ---------------- monotone f32 <-> u32 encoding for atomicMax-based segment max ----
__device__ __forceinline__ unsigned enc_f32(float f) {
  unsigned u = __float_as_uint(f);
  return (u & 0x80000000u) ? ~u : (u | 0x80000000u);
}
__device__ __forceinline__ float dec_f32(unsigned u) {
  return __uint_as_float((u & 0x80000000u) ? (u ^ 0x80000000u) : ~u);
}

// ---------------- zero fill -------------------------------------------------------
__global__ void zero_f32(float* __restrict__ p, long long n) {
  long long i = (long long)blockIdx.x * blockDim.x + threadIdx.x;
  if (i < n) p[i] = 0.0f;
}

// ---------------- WMMA GEMM: Y[N,M] = X[N,K] @ W[M,K]^T  (fp32, 16x16x4) ----------
// One wave per 16x16 output tile. A: lanes0-15 K=k0,k0+1 / lanes16-31 K=k0+2,k0+3.
// B symmetric (N per lane). D: vgpr r -> row r (lanes 0-15) / row r+8 (lanes 16-31).
template <int K>
__global__ void wmma_gemm_xwt(const float* __restrict__ X, const float* __restrict__ W,
                              float* __restrict__ Y, int n_rows, int m_out) {
  int gid = blockIdx.x * blockDim.x + threadIdx.x;
  int wave = gid >> 5;
  int lane = threadIdx.x & 31;
  int tiles_n = m_out >> 4;
  int tile_m = wave / tiles_n;
  int tile_n = wave % tiles_n;
  if (tile_m * 16 >= n_rows) return;  // wave-uniform; EXEC stays all-1s for WMMA

  int l15 = lane & 15;
  int khalf = (lane >> 4) << 1;  // 0 or 2
  const float* xrow = X + (size_t)(tile_m * 16 + l15) * K + khalf;
  const float* wrow = W + (size_t)(tile_n * 16 + l15) * K + khalf;

  v8f acc = {};
#pragma unroll
  for (int k0 = 0; k0 < K; k0 += 4) {
    v2f a, b;
    a.x = xrow[k0]; a.y = xrow[k0 + 1];
    b.x = wrow[k0]; b.y = wrow[k0 + 1];
    acc = __builtin_amdgcn_wmma_f32_16x16x4_f32(false, a, false, b, (short)0, acc,
                                                false, false);
  }

  int out_col = tile_n * 16 + l15;
  int row_base = tile_m * 16 + ((lane >> 4) << 3);
#pragma unroll
  for (int r = 0; r < 8; ++r)
    Y[(size_t)(row_base + r) * m_out + out_col] = acc[r];
}

// ---------------- alpha_src / alpha_dst: wave per (node, head) --------------------
__global__ void alpha_kernel(const float* __restrict__ H, const float* __restrict__ a_src,
                             const float* __restrict__ a_dst, float* __restrict__ as_o,
                             float* __restrict__ ad_o, int n, int heads, int ch) {
  int gid = blockIdx.x * blockDim.x + threadIdx.x;
  int wave = gid >> 5, lane = gid & 31;
  int node = wave / heads, head = wave % heads;
  if (node >= n) return;
  const float* hp = H + ((size_t)node * heads + head) * ch;
  const float* sp = a_src + head * ch;
  const float* dp = a_dst + head * ch;
  float ss = 0.f, sd = 0.f;
  for (int c = lane; c < ch; c += 32) {
    float hv = hp[c];
    ss += hv * sp[c];
    sd += hv * dp[c];
  }
#pragma unroll
  for (int off = 16; off; off >>= 1) {
    ss += __shfl_down(ss, off);
    sd += __shfl_down(sd, off);
  }
  if (lane == 0) {
    as_o[node * heads + head] = ss;
    ad_o[node * heads + head] = sd;
  }
}

__device__ __forceinline__ float leaky(float v) {
  return v > 0.f ? v : v * NEG_SLOPE;
}

// ---------------- edge pass A: segment max ---------------------------------------
__global__ void edge_max(const int* __restrict__ src, const int* __restrict__ dst,
                         const float* __restrict__ as, const float* __restrict__ ad,
                         unsigned* __restrict__ m, int ne, int heads) {
  int e = blockIdx.x * blockDim.x + threadIdx.x;
  if (e >= ne) return;
  int s = src[e], d = dst[e];
  for (int h = 0; h < heads; ++h) {
    float v = leaky(as[s * heads + h] + ad[d * heads + h]);
    atomicMax(&m[d * heads + h], enc_f32(v));
  }
}

// ---------------- edge pass B: sum of exp ----------------------------------------
__global__ void edge_expsum(const int* __restrict__ src, const int* __restrict__ dst,
                            const float* __restrict__ as, const float* __restrict__ ad,
                            const unsigned* __restrict__ m, float* __restrict__ ssum,
                            int ne, int heads) {
  int e = blockIdx.x * blockDim.x + threadIdx.x;
  if (e >= ne) return;
  int s = src[e], d = dst[e];
  for (int h = 0; h < heads; ++h) {
    float v = leaky(as[s * heads + h] + ad[d * heads + h]);
    float mv = dec_f32(m[d * heads + h]);
    if (!isfinite(mv)) mv = 0.f;
    atomicAdd(&ssum[d * heads + h], expf(v - mv));
  }
}

// ---------------- edge pass C: scatter messages (thread = edge x channel) --------
__global__ void edge_scatter(const int* __restrict__ src, const int* __restrict__ dst,
                             const float* __restrict__ as, const float* __restrict__ ad,
                             const unsigned* __restrict__ m, const float* __restrict__ ssum,
                             const float* __restrict__ H, float* __restrict__ out,
                             int ne, int heads, int ch) {
  int gid = blockIdx.x * blockDim.x + threadIdx.x;
  int e = gid / ch, c = gid % ch;
  if (e >= ne) return;
  int s = src[e], d = dst[e];
  for (int h = 0; h < heads; ++h) {
    float v = leaky(as[s * heads + h] + ad[d * heads + h]);
    float mv = dec_f32(m[d * heads + h]);
    if (!isfinite(mv)) mv = 0.f;
    float coef = expf(v - mv) / (ssum[d * heads + h] + 1e-16f);
    atomicAdd(&out[((size_t)d * heads + h) * ch + c],
              H[((size_t)s * heads + h) * ch + c] * coef);
  }
}

// ---------------- bias + relu in place -------------------------------------------
__global__ void bias_relu(float* __restrict__ out, const float* __restrict__ bias,
                          int total, int width) {
  int gid = blockIdx.x * blockDim.x + threadIdx.x;
  if (gid >= total) return;
  float v = out[gid] + bias[gid % width];
  out[gid] = v > 0.f ? v : 0.f;
}

// ---------------- global mean pool -----------------------------------------------
__global__ void pool_sum(const float* __restrict__ h, const int* __restrict__ batch,
                         float* __restrict__ sums, float* __restrict__ cnt, int n, int ch) {
  int gid = blockIdx.x * blockDim.x + threadIdx.x;
  int node = gid / ch, c = gid % ch;
  if (node >= n) return;
  int g = batch[node];
  atomicAdd(&sums[g * ch + c], h[(size_t)node * ch + c]);
  if (c == 0) atomicAdd(&cnt[g], 1.0f);
}

__global__ void pool_div(const float* __restrict__ sums, const float* __restrict__ cnt,
                         float* __restrict__ out, int total, int ch) {
  int gid = blockIdx.x * blockDim.x + threadIdx.x;
  if (gid >= total) return;
  out[gid] = sums[gid] / fmaxf(cnt[gid / ch], 1.0f);
}

// =================================================================================
extern "C" void kernel_launch(void* const* d_in, const int* in_sizes, int n_in,
                              void* d_out, int out_size, void* d_ws, size_t ws_size,
                              hipStream_t stream) {
  const float* x      = (const float*)d_in[0];
  const float* w1     = (const float*)d_in[1];
  const float* a_src1 = (const float*)d_in[2];
  const float* a_dst1 = (const float*)d_in[3];
  const float* b1     = (const float*)d_in[4];
  const float* w2     = (const float*)d_in[5];
  const float* a_src2 = (const float*)d_in[6];
  const float* a_dst2 = (const float*)d_in[7];
  const float* b2     = (const float*)d_in[8];
  const int*   ei     = (const int*)d_in[9];
  const int*   batch  = (const int*)d_in[10];
  const int* src = ei;
  const int* dst = ei + EE;
  float* outp = (float*)d_out;

  // ---- workspace layout (floats) ----
  float* ws = (float*)d_ws;
  size_t o = 0;
  float* h1   = ws + o; o += (size_t)NN * 256;  // layer-1 transformed features
  float* acc1 = ws + o; o += (size_t)NN * 256;  // layer-1 aggregation -> relu(h)
  float* as1  = ws + o; o += (size_t)NN * 4;
  float* ad1  = ws + o; o += (size_t)NN * 4;
  unsigned* m1 = (unsigned*)(ws + o); o += (size_t)NN * 4;
  float* s1v  = ws + o; o += (size_t)NN * 4;
  // layer-2 + pooling alias into h1's region (h1 dead after layer-1 scatter)
  float* h2   = h1;                       // NN*64
  float* acc2 = h1 + (size_t)NN * 64;     // NN*64
  float* as2  = h1 + (size_t)NN * 128;    // NN
  float* ad2  = as2 + NN;                 // NN
  unsigned* m2 = (unsigned*)(ad2 + NN);   // NN
  float* s2v  = (float*)m2 + NN;          // NN
  float* psum = s2v + NN;                 // GG*64
  float* pcnt = psum + GG * 64;           // GG

  const int TB = 256;
  auto blocks = [](long long n, int tb) { return (unsigned)((n + tb - 1) / tb); };

  // ---- layer 1 ----
  zero_f32<<<blocks((long long)NN * 256, TB), TB, 0, stream>>>(acc1, (long long)NN * 256);
  zero_f32<<<blocks((long long)NN * 8, TB), TB, 0, stream>>>((float*)m1, (long long)NN * 8);  // m1+s1v

  // GEMM1: waves = (NN/16)*(256/16) = 100000 ; 8 waves/block
  wmma_gemm_xwt<128><<<100000 * 32 / TB, TB, 0, stream>>>(x, w1, h1, NN, 256);
  alpha_kernel<<<blocks((long long)NN * 4 * 32, TB), TB, 0, stream>>>(h1, a_src1, a_dst1,
                                                                      as1, ad1, NN, 4, 64);
  edge_max<<<blocks(EE, TB), TB, 0, stream>>>(src, dst, as1, ad1, m1, EE, 4);
  edge_expsum<<<blocks(EE, TB), TB, 0, stream>>>(src, dst, as1, ad1, m1, s1v, EE, 4);
  edge_scatter<<<blocks((long long)EE * 64, TB), TB, 0, stream>>>(src, dst, as1, ad1, m1,
                                                                  s1v, h1, acc1, EE, 4, 64);
  bias_relu<<<blocks((long long)NN * 256, TB), TB, 0, stream>>>(acc1, b1, NN * 256, 256);

  // ---- layer 2 (zero acc2..pcnt: contiguous NN*68 + GG*65 floats) ----
  zero_f32<<<blocks((long long)NN * 68 + GG * 65, TB), TB, 0, stream>>>(
      acc2, (long long)NN * 68 + GG * 65);

  // GEMM2: waves = (NN/16)*(64/16) = 25000
  wmma_gemm_xwt<256><<<25000 * 32 / TB, TB, 0, stream>>>(acc1, w2, h2, NN, 64);
  alpha_kernel<<<blocks((long long)NN * 32, TB), TB, 0, stream>>>(h2, a_src2, a_dst2,
                                                                  as2, ad2, NN, 1, 64);
  edge_max<<<blocks(EE, TB), TB, 0, stream>>>(src, dst, as2, ad2, m2, EE, 1);
  edge_expsum<<<blocks(EE, TB), TB, 0, stream>>>(src, dst, as2, ad2, m2, s2v, EE, 1);
  edge_scatter<<<blocks((long long)EE * 64, TB), TB, 0, stream>>>(src, dst, as2, ad2, m2,
                                                                  s2v, h2, acc2, EE, 1, 64);
  bias_relu<<<blocks((long long)NN * 64, TB), TB, 0, stream>>>(acc2, b2, NN * 64, 64);

  // ---- global mean pool ----
  pool_sum<<<blocks((long long)NN * 64, TB), TB, 0, stream>>>(acc2, batch, psum, pcnt, NN, 64);
  pool_div<<<blocks(GG * 64, TB), TB, 0, stream>>>(psum, pcnt, outp, GG * 64, 64);
}